// FCOS_1073741824075
// MI455X (gfx1250) — compile-verified
//
#include <hip/hip_runtime.h>
#include <stdint.h>

// Class-specific NMS, N=8192, IoU 0.5 — CDNA5 (gfx1250, wave32).
//
// Pipeline (all on `stream`):
//   C) 8192x8192 suppression bitmask, 64-col tiles; column boxes staged into LDS
//      via gfx1250 async global->LDS (global_load_async_to_lds_b128 / s_wait_asynccnt)
//   D) sequential suppression scan: ONE wave32, 128 u64 accumulator words held in
//      registers (4/lane), keep-bit broadcast via v_readlane (uniform src lane),
//      global_prefetch_b8 16 rows ahead to hide L2 latency on the serial chain
//   A) max over all box coords (1 block reduction)
//   B) stable descending rank-sort by score + class offset, scatter to sorted arrays
//   E) scatter keep back to original order, write masked [boxes|score] + keep plane

#define NB      8192
#define WORDS   (NB / 64)          // 128 u64 words per mask row
#define IOU_TH  0.5f

typedef unsigned long long u64;

// ---------------- workspace layout (bytes) ----------------
#define OFF_MAXC   0          // 1 float
#define OFF_SIDX   256        // 8192 int        (32768 B)
#define OFF_SBOX   33024      // 8192 float4     (131072 B)
#define OFF_KEEPS  164096     // 8192 uint       (32768 B)
#define OFF_MASK   196864     // 8192*128 u64    (8388608 B)  total ~8.2 MB

// ---------------- C: pairwise suppression bitmask ----------------
// grid = (WORDS, NB/128), block = 128. Each block: 64 column boxes vs 128 rows.
__global__ void nms_mask(const float4* __restrict__ sboxes, u64* __restrict__ mask) {
    __shared__ float4 cbox[64];
    __shared__ float  carea[64];
    const int tid     = threadIdx.x;
    const int colBase = blockIdx.x * 64;
    const int row     = blockIdx.y * 128 + tid;

    // gfx1250 async global->LDS staging of the 64 column boxes (16 B per lane).
    if (tid < 64) {
        uint32_t lds  = (uint32_t)(uintptr_t)&cbox[tid];                 // low 32 bits of flat = LDS addr
        uint64_t gsrc = (uint64_t)(uintptr_t)(sboxes + colBase + tid);
        asm volatile("global_load_async_to_lds_b128 %0, %1, off"
                     :: "v"(lds), "v"(gsrc) : "memory");
    }
    asm volatile("s_wait_asynccnt 0x0" ::: "memory");   // issuing wave: data landed in LDS
    if (tid < 64) {                                     // precompute column areas
        const float4 cb = cbox[tid];
        carea[tid] = (cb.z - cb.x) * (cb.w - cb.y);
    }
    __syncthreads();                                    // publish to the other waves

    const float4 rb = sboxes[row];
    const float rarea = (rb.z - rb.x) * (rb.w - rb.y);
    u64 bits = 0ull;
#pragma unroll 8
    for (int c = 0; c < 64; ++c) {
        const float4 cb = cbox[c];
        const float ix1 = fmaxf(rb.x, cb.x);
        const float iy1 = fmaxf(rb.y, cb.y);
        const float ix2 = fminf(rb.z, cb.z);
        const float iy2 = fminf(rb.w, cb.w);
        const float iw = fmaxf(ix2 - ix1, 0.0f);
        const float ih = fmaxf(iy2 - iy1, 0.0f);
        const float inter = iw * ih;
        const float uni = fmaxf(rarea + carea[c] - inter, 1e-9f);
        const bool sup = (inter > IOU_TH * uni) && ((colBase + c) > row);
        bits |= ((u64)sup) << c;
    }
    mask[(size_t)row * WORDS + blockIdx.x] = bits;
}

// ---------------- D: sequential suppression scan, single wave32 ----------------
__global__ void nms_scan(const u64* __restrict__ mask, unsigned* __restrict__ keepS) {
    const int lane = threadIdx.x;                 // 0..31
    u64 r0 = 0, r1 = 0, r2 = 0, r3 = 0;           // lane owns words lane, lane+32, lane+64, lane+96

    for (int i = 0; i < NB; ++i) {
        // Prefetch row i+16 into WGP cache (gfx1250 global_prefetch_b8).
        const int pi = i + 16;
        if (pi < NB) {
            const u64* p = mask + (size_t)pi * WORDS + lane;
            __builtin_prefetch(p);
            __builtin_prefetch(p + 32);
            __builtin_prefetch(p + 64);
            __builtin_prefetch(p + 96);
        }

        const int w    = i >> 6;       // which u64 word holds bit i
        const int slot = w >> 5;       // which register holds it
        const int src  = w & 31;       // which lane owns it (uniform)
        u64 myword;
        switch (slot) {
            case 0:  myword = r0; break;
            case 1:  myword = r1; break;
            case 2:  myword = r2; break;
            default: myword = r3; break;
        }
        // Uniform-lane broadcast on the serial critical path: v_readlane_b32 x2
        const unsigned lo = (unsigned)__builtin_amdgcn_readlane((int)(myword & 0xffffffffull), src);
        const unsigned hi = (unsigned)__builtin_amdgcn_readlane((int)(myword >> 32), src);
        const u64 word = ((u64)hi << 32) | (u64)lo;
        const bool kp = ((word >> (i & 63)) & 1ull) == 0ull;

        if (kp) {
            const u64* rowp = mask + (size_t)i * WORDS;
            r0 |= rowp[lane];
            r1 |= rowp[lane + 32];
            r2 |= rowp[lane + 64];
            r3 |= rowp[lane + 96];
        }
        if (lane == 0) keepS[i] = kp ? 1u : 0u;
    }
}

// ---------------- A: max coordinate ----------------
__global__ void nms_maxc(const float* __restrict__ boxes, float* __restrict__ maxc) {
    __shared__ float red[1024];
    const int tid = threadIdx.x;
    float m = -3.4e38f;
    for (int j = tid; j < NB * 4; j += 1024) m = fmaxf(m, boxes[j]);
    red[tid] = m;
    __syncthreads();
    for (int s = 512; s > 0; s >>= 1) {
        if (tid < s) red[tid] = fmaxf(red[tid], red[tid + s]);
        __syncthreads();
    }
    if (tid == 0) maxc[0] = red[0];
}

// ---------------- B: stable descending rank + class offset + scatter ----------------
__global__ void nms_rank(const float* __restrict__ boxes,
                         const float* __restrict__ scores,
                         const int* __restrict__ cls,
                         const float* __restrict__ maxc,
                         float4* __restrict__ sboxes,
                         int* __restrict__ sidx) {
    __shared__ float ss[256];
    const int i  = blockIdx.x * 256 + threadIdx.x;
    const float si = scores[i];
    int rank = 0;
    for (int base = 0; base < NB; base += 256) {
        ss[threadIdx.x] = scores[base + threadIdx.x];
        __syncthreads();
#pragma unroll 8
        for (int t = 0; t < 256; ++t) {
            const float sj = ss[t];
            const int j = base + t;
            rank += (int)((sj > si) | ((sj == si) & (j < i)));   // stable: ties -> lower index first
        }
        __syncthreads();
    }
    const float off = (float)cls[i] * (maxc[0] + 1.0f);
    const float4 b = ((const float4*)boxes)[i];
    sboxes[rank] = make_float4(b.x + off, b.y + off, b.z + off, b.w + off);
    sidx[rank]   = i;
}

// ---------------- E: scatter keep, write masked output ----------------
__global__ void nms_out(const float* __restrict__ boxes,
                        const float* __restrict__ scores,
                        const int* __restrict__ sidx,
                        const unsigned* __restrict__ keepS,
                        float* __restrict__ out) {
    const int i = blockIdx.x * 256 + threadIdx.x;  // sorted position
    const int o = sidx[i];                         // original index
    const float k = keepS[i] ? 1.0f : 0.0f;
    const float4 b = ((const float4*)boxes)[o];
    out[o * 5 + 0] = b.x * k;
    out[o * 5 + 1] = b.y * k;
    out[o * 5 + 2] = b.z * k;
    out[o * 5 + 3] = b.w * k;
    out[o * 5 + 4] = scores[o] * k;
    out[5 * NB + o] = k;                           // keep vector as floats
}

extern "C" void kernel_launch(void* const* d_in, const int* in_sizes, int n_in,
                              void* d_out, int out_size, void* d_ws, size_t ws_size,
                              hipStream_t stream) {
    const float* boxes  = (const float*)d_in[0];
    const float* scores = (const float*)d_in[1];
    const int*   cls    = (const int*)d_in[2];

    char* ws = (char*)d_ws;
    float*    maxc   = (float*)(ws + OFF_MAXC);
    int*      sidx   = (int*)(ws + OFF_SIDX);
    float4*   sboxes = (float4*)(ws + OFF_SBOX);
    unsigned* keepS  = (unsigned*)(ws + OFF_KEEPS);
    u64*      mask   = (u64*)(ws + OFF_MASK);

    nms_maxc<<<1, 1024, 0, stream>>>(boxes, maxc);
    nms_rank<<<NB / 256, 256, 0, stream>>>(boxes, scores, cls, maxc, sboxes, sidx);
    dim3 mg(WORDS, NB / 128);
    nms_mask<<<mg, 128, 0, stream>>>(sboxes, mask);
    nms_scan<<<1, 32, 0, stream>>>(mask, keepS);
    nms_out<<<NB / 256, 256, 0, stream>>>(boxes, scores, sidx, keepS, (float*)d_out);
}